// DeltaOrderLoss_57836029608370
// MI455X (gfx1250) — compile-verified
//
#include <hip/hip_runtime.h>
#include <hip/hip_bf16.h>
#include <math.h>

typedef __attribute__((ext_vector_type(2))) float v2f;
typedef __attribute__((ext_vector_type(8))) float v8f;

#define NROWS 512
#define DIM   256
#define NM1   511            // N-1 columns after diagonal drop
#define DELTA_V 0.1f

// f[i] = concat(features[:,0], features[:,1]) row i; features is [256,2,256]
__device__ __forceinline__ const float* frow(const float* feat, int i) {
    return feat + (size_t)(i & 255) * 512 + (size_t)(i >> 8) * 256;
}

// ---------------- kernel 1: row norms |f_i|^2 ----------------
__global__ void norms_kernel(const float* __restrict__ feat, float* __restrict__ nrm) {
    int i = blockIdx.x * blockDim.x + threadIdx.x;
    if (i >= NROWS) return;
    const float* r = frow(feat, i);
    float s = 0.f;
    #pragma unroll 8
    for (int d = 0; d < DIM; ++d) { float v = r[d]; s = fmaf(v, v, s); }
    nrm[i] = s;
}

// ---------------- kernel 2: Gram via WMMA -> distance matrix ----------------
// One wave computes one 16x16 tile of G = F F^T using V_WMMA_F32_16X16X4_F32,
// then fd[r][c] = sqrt(max(|f_r|^2 + |f_c|^2 - 2 G, 0) + (r==c)).
__global__ void gram_dist_kernel(const float* __restrict__ feat,
                                 const float* __restrict__ nrm,
                                 float* __restrict__ fdm) {
    int wave = threadIdx.x >> 5;
    int lane = threadIdx.x & 31;
    int tile = blockIdx.x * 4 + wave;      // 1024 tiles total
    int tr = tile >> 5;                    // 0..31
    int tc = tile & 31;                    // 0..31

    // A 16x4 f32 layout: lanes 0-15 -> M=lane, K={k0,k0+1}; lanes 16-31 -> K={k0+2,k0+3}
    // B 4x16 f32 layout symmetric with N=lane&15.
    int m    = lane & 15;
    int koff = (lane >> 4) << 1;
    const float* arow = frow(feat, tr * 16 + m);
    const float* brow = frow(feat, tc * 16 + m);

    v8f c = {};
    #pragma unroll 4
    for (int k0 = 0; k0 < DIM; k0 += 4) {
        v2f a = *(const v2f*)(arow + k0 + koff);
        v2f b = *(const v2f*)(brow + k0 + koff);
        c = __builtin_amdgcn_wmma_f32_16x16x4_f32(
                /*neg_a=*/false, a, /*neg_b=*/false, b,
                /*c_mod=*/(short)0, c, /*reuse_a=*/false, /*reuse_b=*/false);
    }

    // C/D layout: VGPR v, lanes 0-15 -> M=v, N=lane; lanes 16-31 -> M=v+8, N=lane-16
    int n     = lane & 15;
    int mbase = (lane >> 4) * 8;
    #pragma unroll
    for (int v = 0; v < 8; ++v) {
        int r  = tr * 16 + mbase + v;
        int cc = tc * 16 + n;
        float g  = c[v];
        float sq = nrm[r] + nrm[cc] - 2.0f * g;
        sq = fmaxf(sq, 0.0f);
        if (r == cc) sq += 1.0f;           // eye() guard; diagonal dropped later anyway
        fdm[r * NROWS + cc] = sqrtf(sq);
    }
}

// ---------------- kernel 3: per-row stable counting sort + k-loop ----------------
__global__ void row_loss_kernel(const float* __restrict__ fdm,
                                const int* __restrict__ labels,
                                float* __restrict__ part) {
    __shared__ int   ldcol[NM1];   // label diff per (diag-dropped) column, original order
    __shared__ float sfd[NM1];     // feature dists, sorted by label diff (stable)
    __shared__ int   sld[NM1];     // sorted label diffs
    __shared__ float srk[NM1];     // rank (distinct-group index) per sorted position
    __shared__ int   counts[100];
    __shared__ int   startv[100];
    __shared__ int   rankv[100];
    __shared__ float red[256];

    int row  = blockIdx.x;
    int tid  = threadIdx.x;
    int labi = labels[row & 255];

    for (int v = tid; v < 100; v += 256) counts[v] = 0;
    __syncthreads();

    for (int j = tid; j < NM1; j += 256) {
        int cj = j + (j >= row);                    // skip diagonal
        int d  = labi - labels[cj & 255];
        int v  = d < 0 ? -d : d;                    // |label diff|, < 100
        ldcol[j] = v;
        atomicAdd(&counts[v], 1);
    }
    __syncthreads();

    if (tid == 0) {                                 // exclusive scan + distinct-value ranks
        int run = 0, rk = 0;
        for (int v = 0; v < 100; ++v) {
            startv[v] = run;
            rankv[v]  = rk;
            run += counts[v];
            if (counts[v] > 0) rk++;
        }
    }
    __syncthreads();

    // stable placement: position = group start + #earlier columns with same value
    for (int j = tid; j < NM1; j += 256) {
        int v = ldcol[j];
        int off = 0;
        for (int p = 0; p < j; ++p) off += (ldcol[p] == v);
        int pos = startv[v] + off;
        int cj  = j + (j >= row);
        sfd[pos] = fdm[row * NROWS + cj];
        sld[pos] = v;
        srk[pos] = (float)rankv[v];
    }
    __syncthreads();

    // k in 0..N-3 (510 values); all threads read same LDS word per j -> broadcast
    float acc = 0.f;
    for (int k = tid; k < NROWS - 2; k += 256) {
        int   labk = sld[k];
        float fdk  = sfd[k];
        float rkk  = srk[k];
        float pos = 0.f, neg = 0.f;
        for (int j = 0; j < NM1; ++j) {
            float d  = sfd[j] - fdk;
            float ad = fabsf(d);
            int   lj = sld[j];
            if (lj == labk && j != k && ad > DELTA_V) pos += expf(-ad);
            if (labk < lj) {
                float margin = (srk[j] - rkk) * (1.0f / DELTA_V);
                if (d > -margin) neg += expf(d);
            }
        }
        if (pos == 0.f) pos = 1.f;
        if (neg == 0.f) neg = 1.f;
        acc += logf(neg) - logf(pos);               // == -log(pos/neg)
    }

    red[tid] = acc;
    __syncthreads();
    for (int s = 128; s > 0; s >>= 1) {
        if (tid < s) red[tid] += red[tid + s];
        __syncthreads();
    }
    if (tid == 0) part[row] = red[0];
}

// ---------------- kernel 4: fixed-order final reduction ----------------
__global__ void final_kernel(const float* __restrict__ part, float* __restrict__ out) {
    __shared__ float red[256];
    int tid = threadIdx.x;
    red[tid] = part[tid] + part[tid + 256];
    __syncthreads();
    for (int s = 128; s > 0; s >>= 1) {
        if (tid < s) red[tid] += red[tid + s];
        __syncthreads();
    }
    if (tid == 0) out[0] = red[0] / (512.0f * 511.0f);
}

extern "C" void kernel_launch(void* const* d_in, const int* in_sizes, int n_in,
                              void* d_out, int out_size, void* d_ws, size_t ws_size,
                              hipStream_t stream) {
    const float* feat   = (const float*)d_in[0];   // [256,2,256] f32
    const int*   labels = (const int*)d_in[1];     // [256,1] i32

    float* fdm  = (float*)d_ws;                    // 512*512 f32 distance matrix
    float* nrm  = fdm + NROWS * NROWS;             // 512 f32 row norms
    float* part = nrm + NROWS;                     // 512 f32 per-row partial losses

    norms_kernel    <<<2,     256, 0, stream>>>(feat, nrm);
    gram_dist_kernel<<<256,   128, 0, stream>>>(feat, nrm, fdm);
    row_loss_kernel <<<NROWS, 256, 0, stream>>>(fdm, labels, part);
    final_kernel    <<<1,     256, 0, stream>>>(part, (float*)d_out);
}